// HF_28587302322331
// MI455X (gfx1250) — compile-verified
//
#include <hip/hip_runtime.h>
#include <hip/hip_bf16.h>
#include <stdint.h>

// ---------------------------------------------------------------------------
// Hough voting (PoseCNN-style) for MI455X / gfx1250.
//   label  : (B,H,W) int32          B=2, C=22, H=480, W=640
//   vertex : (B,3C,H,W) f32
//   extents: (C,3) f32
//   meta   : (B,9) f32 (fx at [b,0])
//   out    : (B,C,9) f32
// Workspace layout:
//   acc  : B*C*HW u32 (54 MB, zeroed each call)
//   cnt  : B*C f32    (zeroed each call)
//   zsum : B*C f32    (zeroed each call)
//   pval : B*C*SPLIT u32   (written unconditionally by stage A)
//   pidx : B*C*SPLIT i32   (written unconditionally by stage A)
// ---------------------------------------------------------------------------

#define NB 2
#define NC 22
#define NH 480
#define NW 640
#define NHW (NH * NW)
#define NT_STEPS 32
#define RAY_STEP 4.0f
#define BLK 256
#define SPLIT 16                      // argmax chunks per (b,c) row

#if defined(__gfx1250__) && __has_builtin(__builtin_amdgcn_global_load_async_to_lds_b32)
#define HAVE_ASYNC_LDS 1
#else
#define HAVE_ASYNC_LDS 0
#endif

typedef __attribute__((address_space(1))) int  g_int;
typedef __attribute__((address_space(3))) int  l_int;

__device__ __forceinline__ void wait_asynccnt0() {
#if __has_builtin(__builtin_amdgcn_s_wait_asynccnt)
    __builtin_amdgcn_s_wait_asynccnt(0);
#elif defined(__gfx1250__)
    asm volatile("s_wait_asynccnt 0" ::: "memory");
#endif
}

// ---------------------------------------------------------------------------
// Kernel 1: per-pixel voting.
// One thread per (b, pixel). HW % BLK == 0, so a block never straddles b.
// Labels staged via CDNA5 async global->LDS loads (ASYNCcnt path).
// Per-class cnt/zsum privatized in LDS, flushed once per block.
// ---------------------------------------------------------------------------
__global__ void __launch_bounds__(BLK)
hough_vote_kernel(const int* __restrict__ label,
                  const float* __restrict__ vertex,
                  unsigned* __restrict__ acc,
                  float* __restrict__ cnt,
                  float* __restrict__ zsum)
{
    __shared__ float s_cnt[NC];
    __shared__ float s_z[NC];
    __shared__ int   s_lab[BLK];

    const int tid = threadIdx.x;
    const int gid = blockIdx.x * BLK + tid;
    const int b   = gid / NHW;
    const int p   = gid - b * NHW;

    if (tid < NC) { s_cnt[tid] = 0.0f; s_z[tid] = 0.0f; }

#if HAVE_ASYNC_LDS
    // CDNA5 async copy: copies label[gid] from global memory into s_lab[tid] in LDS; tracked by ASYNCcnt.
    __builtin_amdgcn_global_load_async_to_lds_b32(
        (g_int*)(label + gid),
        (l_int*)&s_lab[tid],
        0, 0);
    wait_asynccnt0();
#else
    s_lab[tid] = label[gid];
#endif
    __syncthreads();

    const int lab = s_lab[tid];

    // Gather direction + depth for this pixel's class: channel = 3*lab + k.
    const float* vb = vertex + (((size_t)b * (3 * NC) + 3 * lab) * (size_t)NHW + p);
    const float nx0 = vb[0];
    const float ny0 = vb[(size_t)NHW];
    const float z   = vb[2 * (size_t)NHW];

    atomicAdd(&s_cnt[lab], 1.0f);
    atomicAdd(&s_z[lab], z);

    if (lab > 0) {
        // inv_norm = 1/(sqrt(nx^2+ny^2)+1e-8)   (unfused, like the reference)
        const float n2  = __fadd_rn(__fmul_rn(nx0, nx0), __fmul_rn(ny0, ny0));
        const float inv = 1.0f / (sqrtf(n2) + 1e-8f);
        const float nx  = __fmul_rn(nx0, inv);
        const float ny  = __fmul_rn(ny0, inv);

        const float xs = (float)(p % NW);
        const float ys = (float)(p / NW);

        unsigned* abase = acc + ((size_t)b * NC + lab) * (size_t)NHW;

#pragma unroll
        for (int ti = 1; ti <= NT_STEPS; ++ti) {
            const float t  = (float)ti * RAY_STEP;
            // round-half-to-even == jnp.round; keep mul/add unfused.
            const int   px = (int)rintf(__fadd_rn(xs, __fmul_rn(nx, t)));
            const int   py = (int)rintf(__fadd_rn(ys, __fmul_rn(ny, t)));
            if (px >= 0 && px < NW && py >= 0 && py < NH) {
                atomicAdd(abase + (py * NW + px), 1u);
            }
        }
    }

    __syncthreads();
    if (tid < NC && s_cnt[tid] != 0.0f) {
        atomicAdd(&cnt[b * NC + tid],  s_cnt[tid]);
        atomicAdd(&zsum[b * NC + tid], s_z[tid]);
    }
}

// ---------------------------------------------------------------------------
// Kernel 2a: partial argmax. Grid = NB*NC*SPLIT blocks; each block scans
// HW/SPLIT elements of one (b,c) row (uint4-vectorized) and writes one
// (max, first-index) partial. First-index tie-break == jnp.argmax.
// ---------------------------------------------------------------------------
__global__ void __launch_bounds__(BLK)
hough_argmax_partial_kernel(const unsigned* __restrict__ acc,
                            unsigned* __restrict__ pval,
                            int* __restrict__ pidx)
{
    const int gb    = blockIdx.x;          // bc * SPLIT + chunk
    const int bc    = gb / SPLIT;
    const int chunk = gb - bc * SPLIT;
    const int tid   = threadIdx.x;

    const int Q4     = NHW / 4;            // 76800 uint4 per row
    const int CHUNK4 = Q4 / SPLIT;         // 4800 uint4 per chunk
    const int base4  = chunk * CHUNK4;

    const uint4* a4 = (const uint4*)(acc + (size_t)bc * (size_t)NHW);

    unsigned best = 0u;
    int      bidx = 0x7FFFFFFF;

    for (int i = base4 + tid; i < base4 + CHUNK4; i += BLK) {
        const uint4 v = a4[i];
        const unsigned vals[4] = { v.x, v.y, v.z, v.w };
#pragma unroll
        for (int j = 0; j < 4; ++j) {
            const int idx = 4 * i + j;     // global index within the row
            if (vals[j] > best || (vals[j] == best && idx < bidx)) {
                best = vals[j];
                bidx = idx;
            }
        }
    }

    __shared__ unsigned sv[BLK];
    __shared__ int      si[BLK];
    sv[tid] = best;
    si[tid] = bidx;
    __syncthreads();

#pragma unroll
    for (int s = BLK / 2; s > 0; s >>= 1) {
        if (tid < s) {
            const unsigned ov = sv[tid + s];
            const int      oi = si[tid + s];
            if (ov > sv[tid] || (ov == sv[tid] && oi < si[tid])) {
                sv[tid] = ov;
                si[tid] = oi;
            }
        }
        __syncthreads();
    }

    if (tid == 0) {
        pval[gb] = sv[0];
        pidx[gb] = si[0];
    }
}

// ---------------------------------------------------------------------------
// Kernel 2b: combine SPLIT partials per (b,c) and emit the 9-float row.
// Grid = NB*NC blocks of SPLIT*2 threads (only lane work matters; tiny).
// ---------------------------------------------------------------------------
__global__ void __launch_bounds__(32)
hough_finalize_kernel(const unsigned* __restrict__ pval,
                      const int* __restrict__ pidx,
                      const float* __restrict__ cnt,
                      const float* __restrict__ zsum,
                      const float* __restrict__ extents,
                      const float* __restrict__ meta,
                      float* __restrict__ out)
{
    const int bc = blockIdx.x;             // b*NC + c
    const int b  = bc / NC;
    const int c  = bc - b * NC;

    if (threadIdx.x == 0) {
        unsigned best = 0u;
        int      peak = 0x7FFFFFFF;
        // Ascending chunk order => ascending global index => first-index
        // tie-break matches jnp.argmax exactly.
#pragma unroll
        for (int k = 0; k < SPLIT; ++k) {
            const unsigned v = pval[bc * SPLIT + k];
            const int      i = pidx[bc * SPLIT + k];
            if (v > best || (v == best && i < peak)) { best = v; peak = i; }
        }

        const float votes = (float)best;
        const float cx = (float)(peak % NW);
        const float cy = (float)(peak / NW);

        const float cn = cnt[bc];
        const float zs = zsum[bc];
        const float depth = zs / fmaxf(cn, 1.0f);

        const float fx = meta[b * 9 + 0];
        const float e0 = extents[c * 3 + 0];
        const float e1 = extents[c * 3 + 1];
        const float e2 = extents[c * 3 + 2];
        const float diag = sqrtf(e0 * e0 + e1 * e1 + e2 * e2);
        const float half = 0.5f * diag * fx / fmaxf(fabsf(depth), 0.001f);

        float* o = out + (size_t)bc * 9;
        o[0] = (float)c;
        o[1] = votes;
        o[2] = cx - half;
        o[3] = cy - half;
        o[4] = cx + half;
        o[5] = cy + half;
        o[6] = cx;
        o[7] = cy;
        o[8] = depth;
    }
}

// ---------------------------------------------------------------------------
extern "C" void kernel_launch(void* const* d_in, const int* in_sizes, int n_in,
                              void* d_out, int out_size, void* d_ws, size_t ws_size,
                              hipStream_t stream) {
    const int*   label   = (const int*)d_in[0];
    const float* vertex  = (const float*)d_in[1];
    const float* extents = (const float*)d_in[2];
    const float* meta    = (const float*)d_in[3];
    // d_in[4] = gt (unused), d_in[5] = is_train (unused)

    const size_t accElems = (size_t)NB * NC * NHW;          // 13,516,800
    const size_t accBytes = accElems * sizeof(unsigned);    // ~54 MB
    unsigned* acc  = (unsigned*)d_ws;
    float*    cnt  = (float*)((char*)d_ws + accBytes);
    float*    zsum = cnt + NB * NC;
    unsigned* pval = (unsigned*)(zsum + NB * NC);
    int*      pidx = (int*)(pval + NB * NC * SPLIT);

    // Zero accumulator + stats every call (graph-capture-safe memset node).
    // pval/pidx are written unconditionally by stage A; no zeroing needed.
    (void)hipMemsetAsync(d_ws, 0, accBytes + 2 * (size_t)NB * NC * sizeof(float), stream);

    const int voteBlocks = (NB * NHW) / BLK;                // 2400, no straddle
    hough_vote_kernel<<<voteBlocks, BLK, 0, stream>>>(label, vertex, acc, cnt, zsum);

    hough_argmax_partial_kernel<<<NB * NC * SPLIT, BLK, 0, stream>>>(acc, pval, pidx);

    hough_finalize_kernel<<<NB * NC, 32, 0, stream>>>(pval, pidx, cnt, zsum,
                                                      extents, meta, (float*)d_out);
}